// DHMRepairModule_44607530336275
// MI455X (gfx1250) — compile-verified
//
#include <hip/hip_runtime.h>
#include <hip/hip_bf16.h>
#include <math.h>

#define NTHREADS   256
#define C_CH       256
#define FHW        14
#define PIX        196                    // 14*14
#define STAGE_CH   128
#define STAGE_FLOATS (STAGE_CH * PIX)     // 25088 floats = 100352 B
#define A_STRIDE   197                    // 196 + 1 pad -> conflict-free banks
#define A_FLOATS   (16 * A_STRIDE)        // 3152 floats
#define OUT_STRIDE 1298                   // 4 + 1 + 4 + 9 + 5*256
#define FEAT_OFF   18
#define PAD_WF     1024.0f
#define PAD_HF     1024.0f

typedef __attribute__((ext_vector_type(2))) float v2f;
typedef __attribute__((ext_vector_type(8))) float v8f;
typedef __attribute__((ext_vector_type(4))) int   v4i;

// Address-space-qualified vector typedefs (attribute attaches to pointee type).
typedef __attribute__((address_space(1))) v4i v4i_glob;
typedef __attribute__((address_space(3))) v4i v4i_lds;

#if defined(__has_builtin)
#  if __has_builtin(__builtin_amdgcn_wmma_f32_16x16x4_f32)
#    define HAVE_WMMA_F32_16X16X4 1
#  endif
#  if __has_builtin(__builtin_amdgcn_global_load_async_to_lds_b128)
#    define HAVE_ASYNC_LDS 1
#  endif
#  if __has_builtin(__builtin_amdgcn_s_wait_asynccnt)
#    define HAVE_WAIT_ASYNC 1
#  endif
#endif

// ---------------------------------------------------------------- box helpers
struct B4 { float x1, y1, x2, y2; };

__device__ __forceinline__ B4 box_from_residual(const B4& g, const float* r, float s) {
  float w = fmaxf(g.x2 - g.x1, 1.0f);
  float h = fmaxf(g.y2 - g.y1, 1.0f);
  float r0 = g.x1 - r[0] * w * s;
  float r1 = g.y1 - r[1] * h * s;
  float r2 = g.x2 - r[2] * w * s;
  float r3 = g.y2 - r[3] * h * s;
  B4 o;
  o.x1 = fminf(r0, r2 - 1.0f);
  o.y1 = fminf(r1, r3 - 1.0f);
  o.x2 = fmaxf(r2, o.x1 + 1.0f);
  o.y2 = fmaxf(r3, o.y1 + 1.0f);
  return o;
}

__device__ __forceinline__ float aligned_iou(const B4& a, const B4& b) {
  float xx1 = fmaxf(a.x1, b.x1), yy1 = fmaxf(a.y1, b.y1);
  float xx2 = fminf(a.x2, b.x2), yy2 = fminf(a.y2, b.y2);
  float inter = fmaxf(xx2 - xx1, 0.f) * fmaxf(yy2 - yy1, 0.f);
  float a1 = fmaxf(a.x2 - a.x1, 0.f) * fmaxf(a.y2 - a.y1, 0.f);
  float a2 = fmaxf(b.x2 - b.x1, 0.f) * fmaxf(b.y2 - b.y1, 0.f);
  float u  = fmaxf(a1 + a2 - inter, 1.1920929e-7f);
  return inter / u;
}

__device__ __forceinline__ B4 sanitize_box(const B4& b) {
  B4 o;
  o.x1 = fminf(b.x1, b.x2); o.y1 = fminf(b.y1, b.y2);
  o.x2 = fmaxf(b.x1, b.x2); o.y2 = fmaxf(b.y1, b.y2);
  o.x2 = fmaxf(o.x2, o.x1 + 1.f);
  o.y2 = fmaxf(o.y2, o.y1 + 1.f);
  return o;
}

__device__ __forceinline__ B4 clip_box(const B4& b) {
  B4 o;
  o.x1 = fminf(fmaxf(b.x1, 0.f), PAD_WF - 1.f);
  o.y1 = fminf(fmaxf(b.y1, 0.f), PAD_HF - 1.f);
  o.x2 = fminf(fmaxf(b.x2, 0.f), PAD_WF - 1.f);
  o.y2 = fminf(fmaxf(b.y2, 0.f), PAD_HF - 1.f);
  o.x2 = fminf(fmaxf(o.x2, o.x1 + 1.f), PAD_WF);
  o.y2 = fminf(fmaxf(o.y2, o.y1 + 1.f), PAD_HF);
  return o;
}

// ---------------------------------------------------------------- async stage
__device__ __forceinline__ void issue_stage(const float* g, float* l, int tid) {
  const int nchunks = STAGE_FLOATS / 4;   // 6272 x 16B, fully coalesced
  for (int i = tid; i < nchunks; i += NTHREADS) {
#if defined(HAVE_ASYNC_LDS)
    __builtin_amdgcn_global_load_async_to_lds_b128(
        (v4i_glob*)(g + i * 4), (v4i_lds*)(l + i * 4), 0, 0);
#else
    float4 v = *(const float4*)(g + i * 4);
    *(float4*)(l + i * 4) = v;
#endif
  }
}

__device__ __forceinline__ void wait_async_all() {
#if defined(HAVE_ASYNC_LDS)
#  if defined(HAVE_WAIT_ASYNC)
  __builtin_amdgcn_s_wait_asynccnt(0);
#  else
  asm volatile("s_wait_asynccnt 0" ::: "memory");
#  endif
#endif
  __syncthreads();
}

// ---------------------------------------------------------------- main kernel
__global__ void __launch_bounds__(NTHREADS)
dhm_repair_kernel(const float* __restrict__ boxes, const float* __restrict__ deltas,
                  const float* __restrict__ gt,    const float* __restrict__ res,
                  const float* __restrict__ cls,   const float* __restrict__ ctr,
                  const float* __restrict__ fm,    const int*   __restrict__ lvl,
                  float* __restrict__ out, int N) {
  extern __shared__ float smem[];
  float* FMs = smem;                       // STAGE_FLOATS
  float* A   = smem + STAGE_FLOATS;        // A_FLOATS   (16 x 197)
  float* bbL = A + A_FLOATS;               // 4 floats: border box

  const int box = blockIdx.x;
  if (box >= N) return;
  const int tid = threadIdx.x;
  const float* gfm = fm + (size_t)box * (C_CH * PIX);
  float* obase = out + (size_t)box * OUT_STRIDE;

  // kick off async DMA of first 128 channels into LDS
  issue_stage(gfm, FMs, tid);

  // zero the weight matrix while the DMA flies
  for (int i = tid; i < A_FLOATS; i += NTHREADS) A[i] = 0.f;

  if (tid == 0) {
    // ---- replay scan
    B4 g = { gt[box*4+0], gt[box*4+1], gt[box*4+2], gt[box*4+3] };
    float rr[4] = { res[box*4+0], res[box*4+1], res[box*4+2], res[box*4+3] };
    float scale = 1.0f;
    B4 rb = box_from_residual(g, rr, scale);
    float riou = aligned_iou(rb, g);
    #pragma unroll
    for (int it = 0; it < 8; ++it) {
      bool act = (riou >= 0.5f) && (scale < 4.0f);
      float ns = act ? fminf(4.0f, scale * 1.25f) : scale;
      B4 cand = box_from_residual(g, rr, ns);
      float ciou = aligned_iou(cand, g);
      if (act) { rb = cand; riou = ciou; }
      scale = ns;
    }
    // ---- refined box
    B4 b0 = { boxes[box*4+0], boxes[box*4+1], boxes[box*4+2], boxes[box*4+3] };
    float w = fmaxf(b0.x2 - b0.x1, 1.f), h = fmaxf(b0.y2 - b0.y1, 1.f);
    B4 nb = { b0.x1 + deltas[box*4+0]*w, b0.y1 + deltas[box*4+1]*h,
              b0.x2 + deltas[box*4+2]*w, b0.y2 + deltas[box*4+3]*h };
    B4 ref = clip_box(sanitize_box(nb));
    // ---- emit scalar outputs
    obase[0] = rb.x1; obase[1] = rb.y1; obase[2] = rb.x2; obase[3] = rb.y2;
    obase[4] = riou;
    obase[5] = ref.x1; obase[6] = ref.y1; obase[7] = ref.x2; obase[8] = ref.y2;
    float bw = fmaxf(ref.x2 - ref.x1, 1.f), bh = fmaxf(ref.y2 - ref.y1, 1.f);
    obase[9]  = (ref.x1 + ref.x2) * 0.5f / PAD_WF;
    obase[10] = (ref.y1 + ref.y2) * 0.5f / PAD_HF;
    obase[11] = bw / PAD_WF;
    obase[12] = bh / PAD_HF;
    obase[13] = logf(fmaxf(bw / bh, 1e-6f));
    obase[14] = bw * bh / (PAD_WF * PAD_HF);
    obase[15] = cls[box];
    obase[16] = ctr[box];
    obase[17] = (float)lvl[box] * 0.25f;
    // border-sampling box (reference re-sanitizes + re-clips)
    B4 pb = clip_box(sanitize_box(ref));
    bbL[0] = pb.x1; bbL[1] = pb.y1; bbL[2] = pb.x2; bbL[3] = pb.y2;
  }
  __syncthreads();

  // ---- scatter bilinear weights of the 21 border points into A (ds_add_f32)
  if (tid < 21) {
    float x1 = bbL[0], y1 = bbL[1], x2 = bbL[2], y2 = bbL[3];
    float px, py, sc; int row;
    if (tid == 0) { px = (x1 + x2) * 0.5f; py = (y1 + y2) * 0.5f; row = 0; sc = 1.0f; }
    else {
      int side = (tid - 1) / 5;
      float t = (float)((tid - 1) % 5) * 0.25f;
      row = side + 1; sc = 0.2f;
      if      (side == 0) { px = x1;                    py = y1 + (y2 - y1) * t; }
      else if (side == 1) { px = x1 + (x2 - x1) * t;    py = y1;                 }
      else if (side == 2) { px = x2;                    py = y1 + (y2 - y1) * t; }
      else                { px = x1 + (x2 - x1) * t;    py = y2;                 }
    }
    float fx = fminf(fmaxf(px, 0.f), PAD_WF - 1.f) * ((float)(FHW - 1) / (PAD_WF - 1.f));
    float fy = fminf(fmaxf(py, 0.f), PAD_HF - 1.f) * ((float)(FHW - 1) / (PAD_HF - 1.f));
    float x0f = floorf(fx), y0f = floorf(fy);
    float wx = fx - x0f, wy = fy - y0f;
    int ix0 = (int)fminf(fmaxf(x0f,       0.f), 13.f);
    int ix1 = (int)fminf(fmaxf(x0f + 1.f, 0.f), 13.f);
    int iy0 = (int)fminf(fmaxf(y0f,       0.f), 13.f);
    int iy1 = (int)fminf(fmaxf(y0f + 1.f, 0.f), 13.f);
    float* Ar = A + row * A_STRIDE;
    atomicAdd(&Ar[iy0 * FHW + ix0], sc * (1.f - wx) * (1.f - wy));
    atomicAdd(&Ar[iy0 * FHW + ix1], sc * wx * (1.f - wy));
    atomicAdd(&Ar[iy1 * FHW + ix0], sc * (1.f - wx) * wy);
    atomicAdd(&Ar[iy1 * FHW + ix1], sc * wx * wy);
  }
  wait_async_all();   // stage-0 DMA complete + weight scatter barrier

  // ---- per-wave 16-channel tile GEMM: feats = A(16x196) x FM(196x16tile)
  const int wave = tid >> 5;
  const int lane = tid & 31;
  const int nIdx = lane & 15;
  const int kg   = (lane >> 4) << 1;     // K sub-offset 0 / 2 (16x16x4 layout)

  for (int s = 0; s < 2; ++s) {
    if (s == 1) {
      __syncthreads();                   // everyone done reading stage 0
      issue_stage(gfm + STAGE_FLOATS, FMs, tid);
      wait_async_all();
    }
    const float* Ap = A   + nIdx * A_STRIDE + kg;           // A row = lane%16
    const float* Bp = FMs + (wave * 16 + nIdx) * PIX + kg;  // B col = channel
    v8f acc = {0.f, 0.f, 0.f, 0.f, 0.f, 0.f, 0.f, 0.f};
#if defined(HAVE_WMMA_F32_16X16X4)
    #pragma unroll 7
    for (int k = 0; k < PIX; k += 4) {
      v2f a; a.x = Ap[k]; a.y = Ap[k + 1];
      v2f b; b.x = Bp[k]; b.y = Bp[k + 1];
      acc = __builtin_amdgcn_wmma_f32_16x16x4_f32(
          false, a, false, b, (short)0, acc, false, false);
    }
#else
    // VALU fallback (lanes duplicate across half-waves; only lanes<16 store)
    const float* Bc = FMs + (wave * 16 + nIdx) * PIX;
    float f0 = 0.f, f1 = 0.f, f2 = 0.f, f3 = 0.f, f4 = 0.f;
    for (int k = 0; k < PIX; ++k) {
      float v = Bc[k];
      f0 = fmaf(A[0 * A_STRIDE + k], v, f0);
      f1 = fmaf(A[1 * A_STRIDE + k], v, f1);
      f2 = fmaf(A[2 * A_STRIDE + k], v, f2);
      f3 = fmaf(A[3 * A_STRIDE + k], v, f3);
      f4 = fmaf(A[4 * A_STRIDE + k], v, f4);
    }
    acc[0] = f0; acc[1] = f1; acc[2] = f2; acc[3] = f3; acc[4] = f4;
#endif
    if (lane < 16) {     // D rows 0..4 live in VGPRs 0..4, lanes 0..15
      int ch = s * STAGE_CH + wave * 16 + lane;
      float* o = obase + FEAT_OFF + ch;
      o[0]        = acc[0];   // center
      o[C_CH]     = acc[1];   // left mean
      o[2 * C_CH] = acc[2];   // top mean
      o[3 * C_CH] = acc[3];   // right mean
      o[4 * C_CH] = acc[4];   // bottom mean
    }
  }
}

extern "C" void kernel_launch(void* const* d_in, const int* in_sizes, int n_in,
                              void* d_out, int out_size, void* d_ws, size_t ws_size,
                              hipStream_t stream) {
  const float* boxes  = (const float*)d_in[0];
  const float* deltas = (const float*)d_in[1];
  const float* gt     = (const float*)d_in[2];
  const float* res    = (const float*)d_in[3];
  const float* cls    = (const float*)d_in[4];
  const float* ctr    = (const float*)d_in[5];
  const float* fm     = (const float*)d_in[6];
  const int*   lvl    = (const int*)d_in[7];
  float* out = (float*)d_out;
  int N = in_sizes[0] / 4;
  size_t shmem = (size_t)(STAGE_FLOATS + A_FLOATS + 8) * sizeof(float); // ~113 KB
  dhm_repair_kernel<<<N, NTHREADS, shmem, stream>>>(boxes, deltas, gt, res, cls, ctr,
                                                    fm, lvl, out, N);
}